// DepthMask2PointCloudUltra_57243324121545
// MI455X (gfx1250) — compile-verified
//
#include <hip/hip_runtime.h>
#include <stdint.h>

namespace {
constexpr int kH = 150;
constexpr int kW = 200;
constexpr int kN = kH * kW;                   // 30000 pixels
constexpr int kNW = kN / 4;                   // 7500 packed tag words
constexpr int kK = 1024;                      // max points per person
constexpr int kP = 5;                         // persons 1..5
constexpr int kBT = 1024;                     // 32 wave32 waves per workgroup
constexpr int kChunk = 32;                    // raster chunk per thread (938 active)
constexpr int kStride = kP * (kK + 1);        // 5125 = per-channel output stride
constexpr uint32_t kDepthMinBits = 0x40400000u;  // bits of 3.0f (valid: d > 3.0)
}

typedef unsigned int v4u __attribute__((ext_vector_type(4)));
typedef int v8i __attribute__((ext_vector_type(8)));
typedef int v4i __attribute__((ext_vector_type(4)));

__global__ __launch_bounds__(kBT)
void DepthMask2PointCloudUltra_57243324121545_kernel(const float* __restrict__ in,
                                                     float* __restrict__ out)
{
  __shared__ __align__(16) uint32_t dbits[kN];   // depth plane bits (120 KB), TDM target
  __shared__ __align__(16) uint32_t pidw[kNW];   // 4 person tags packed per word (30 KB)
  __shared__ uint32_t hist[256];                 // radix-select histogram
  __shared__ uint32_t scan[kBT];                 // block scan / reduction scratch
  __shared__ uint32_t shk;                       // remaining rank during radix select
  __shared__ uint32_t shbin;                     // selected bin per radix pass
  __shared__ uint32_t nv[kP + 1];                // valid count per person
  __shared__ float    lbA[kP + 1];               // IQR lower bound per person
  __shared__ float    ubA[kP + 1];               // IQR upper bound per person

  const int b   = blockIdx.x;
  const int tid = threadIdx.x;
  const float* dplane = in + (size_t)b * 3 * kN;   // channel 0: depth
  const float* iplane = dplane + kN;               // channel 1: indicator
  float* ob = out + (size_t)b * 3 * kStride;

  const uint4* dbits4 = (const uint4*)dbits;
  const uint8_t* pid8 = (const uint8_t*)pidw;

  // ---- Phase 0: TDM async DMA of the depth plane (120 KB) global -> LDS -------
  if (tid < 32) {
    uint64_t ga  = (uint64_t)(uintptr_t)dplane;
    uint32_t lds = (uint32_t)(uintptr_t)&dbits[0];  // flat addr low 32 bits == LDS offset
    v4u g0;
    g0.x = 1u;                                       // count=1, user descriptor
    g0.y = lds;                                      // lds_addr
    g0.z = (uint32_t)ga;                             // global_addr[31:0]
    g0.w = ((uint32_t)(ga >> 32) & 0x01FFFFFFu) | (2u << 30);  // addr[56:32] | type=2
    v8i g1;
    g1[0] = (int)(2u << 16);                                      // data_size=4B, mask=0
    g1[1] = (int)(((uint32_t)kN & 0xFFFFu) << 16);                // tensor_dim0[15:0]
    g1[2] = (int)((((uint32_t)kN >> 16) & 0xFFFFu) | (1u << 16)); // dim0 hi | tensor_dim1=1
    g1[3] = (int)((uint32_t)kN << 16);                            // tile_dim0 = 30000
    g1[4] = 0;
    g1[5] = (int)kN;                                              // tensor_dim0_stride
    g1[6] = 0;
    g1[7] = 0;
    v4i gz = {0, 0, 0, 0};
#if __clang_major__ >= 23
    v8i gz8 = {0, 0, 0, 0, 0, 0, 0, 0};
    __builtin_amdgcn_tensor_load_to_lds(g0, g1, gz, gz, gz8, 0);
#else
    __builtin_amdgcn_tensor_load_to_lds(g0, g1, gz, gz, 0);
#endif
  }

  // Overlapped with the DMA: zero output slice, build packed person tags (f32x4 loads).
  for (int i = tid; i < 3 * kStride; i += kBT) ob[i] = 0.0f;
  {
    const float4* ip4 = (const float4*)iplane;
    for (int w = tid; w < kNW; w += kBT) {
      float4 f = ip4[w];
      int p0 = (int)rintf(f.x), p1 = (int)rintf(f.y);
      int p2 = (int)rintf(f.z), p3 = (int)rintf(f.w);
      uint32_t t0 = (p0 >= 1 && p0 <= kP) ? (uint32_t)p0 : 0u;
      uint32_t t1 = (p1 >= 1 && p1 <= kP) ? (uint32_t)p1 : 0u;
      uint32_t t2 = (p2 >= 1 && p2 <= kP) ? (uint32_t)p2 : 0u;
      uint32_t t3 = (p3 >= 1 && p3 <= kP) ? (uint32_t)p3 : 0u;
      pidw[w] = t0 | (t1 << 8) | (t2 << 16) | (t3 << 24);
    }
  }
  if (tid < 32) __builtin_amdgcn_s_wait_tensorcnt(0);
  __syncthreads();

  // ---- Phase 1: fused validity fixup (d > 3.0) + per-person valid counts ------
  auto blockSum = [&](uint32_t c) -> uint32_t {
    scan[tid] = c;
    __syncthreads();
    for (int off = kBT >> 1; off > 0; off >>= 1) {
      if (tid < off) scan[tid] += scan[tid + off];
      __syncthreads();
    }
    uint32_t r = scan[0];
    __syncthreads();
    return r;
  };
  {
    uint32_t c1 = 0, c2 = 0, c3 = 0, c4 = 0, c5 = 0;
    for (int w = tid; w < kNW; w += kBT) {
      uint32_t tags = pidw[w];
      if (tags) {
        uint4 d4 = dbits4[w];
        if ((tags & 0x000000FFu) && d4.x <= kDepthMinBits) tags &= 0xFFFFFF00u;
        if ((tags & 0x0000FF00u) && d4.y <= kDepthMinBits) tags &= 0xFFFF00FFu;
        if ((tags & 0x00FF0000u) && d4.z <= kDepthMinBits) tags &= 0xFF00FFFFu;
        if ((tags & 0xFF000000u) && d4.w <= kDepthMinBits) tags &= 0x00FFFFFFu;
        pidw[w] = tags;
        uint32_t b0 = tags & 0xFFu, b1 = (tags >> 8) & 0xFFu;
        uint32_t b2 = (tags >> 16) & 0xFFu, b3 = tags >> 24;
        c1 += (b0 == 1) + (b1 == 1) + (b2 == 1) + (b3 == 1);
        c2 += (b0 == 2) + (b1 == 2) + (b2 == 2) + (b3 == 2);
        c3 += (b0 == 3) + (b1 == 3) + (b2 == 3) + (b3 == 3);
        c4 += (b0 == 4) + (b1 == 4) + (b2 == 4) + (b3 == 4);
        c5 += (b0 == 5) + (b1 == 5) + (b2 == 5) + (b3 == 5);
      }
    }
    uint32_t r12 = blockSum(c1 | (c2 << 16));
    uint32_t r34 = blockSum(c3 | (c4 << 16));
    uint32_t r5  = blockSum(c5);
    if (tid == 0) {
      nv[1] = r12 & 0xFFFFu; nv[2] = r12 >> 16;
      nv[3] = r34 & 0xFFFFu; nv[4] = r34 >> 16;
      nv[5] = r5;
    }
    __syncthreads();
  }

  // ---- Phase 2: exact k-th order statistic via 4-pass MSB radix select --------
  // Valid depths are positive floats -> uint32 bits preserve ordering.
  auto selectK = [&](uint32_t p, uint32_t k) -> uint32_t {
    uint32_t prefix = 0, pmask = 0;
    if (tid == 0) shk = k;
    __syncthreads();
    for (int shift = 24; shift >= 0; shift -= 8) {
      for (int bi = tid; bi < 256; bi += kBT) hist[bi] = 0;
      __syncthreads();
      for (int w = tid; w < kNW; w += kBT) {
        uint32_t tags = pidw[w];
        uint32_t b0 = tags & 0xFFu, b1 = (tags >> 8) & 0xFFu;
        uint32_t b2 = (tags >> 16) & 0xFFu, b3 = tags >> 24;
        if ((b0 == p) | (b1 == p) | (b2 == p) | (b3 == p)) {
          uint4 d4 = dbits4[w];
          if (b0 == p && (d4.x & pmask) == prefix) atomicAdd(&hist[(d4.x >> shift) & 255u], 1u);
          if (b1 == p && (d4.y & pmask) == prefix) atomicAdd(&hist[(d4.y >> shift) & 255u], 1u);
          if (b2 == p && (d4.z & pmask) == prefix) atomicAdd(&hist[(d4.z >> shift) & 255u], 1u);
          if (b3 == p && (d4.w & pmask) == prefix) atomicAdd(&hist[(d4.w >> shift) & 255u], 1u);
        }
      }
      __syncthreads();
      uint32_t kk = shk;      // uniform snapshot of remaining rank
      __syncthreads();        // all reads complete before the selected lane rewrites shk
      if (tid < 256) scan[tid] = hist[tid];
      __syncthreads();
      for (int off = 1; off < 256; off <<= 1) {   // parallel 256-bin inclusive scan
        uint32_t v = 0, a = 0;
        if (tid < 256) { v = scan[tid]; a = (tid >= off) ? scan[tid - off] : 0u; }
        __syncthreads();
        if (tid < 256) scan[tid] = v + a;
        __syncthreads();
      }
      if (tid < 256) {
        uint32_t incl = scan[tid];
        uint32_t excl = incl - hist[tid];
        if (kk >= excl && kk < incl) { shbin = (uint32_t)tid; shk = kk - excl; }
      }
      __syncthreads();
      prefix |= (shbin << shift);
      pmask  |= (255u << shift);
      __syncthreads();
    }
    return prefix;   // exact bit pattern of the k-th smallest valid depth
  };

  // One-pass probe: count(<= v) and min(> v) among person p's valid depths.
  auto rankProbe = [&](uint32_t p, uint32_t vbits, uint32_t* cntLE, uint32_t* minAbove) {
    uint32_t cle = 0, mab = 0xFFFFFFFFu;
    for (int w = tid; w < kNW; w += kBT) {
      uint32_t tags = pidw[w];
      uint32_t b0 = tags & 0xFFu, b1 = (tags >> 8) & 0xFFu;
      uint32_t b2 = (tags >> 16) & 0xFFu, b3 = tags >> 24;
      if ((b0 == p) | (b1 == p) | (b2 == p) | (b3 == p)) {
        uint4 d4 = dbits4[w];
        if (b0 == p) { if (d4.x <= vbits) cle++; else if (d4.x < mab) mab = d4.x; }
        if (b1 == p) { if (d4.y <= vbits) cle++; else if (d4.y < mab) mab = d4.y; }
        if (b2 == p) { if (d4.z <= vbits) cle++; else if (d4.z < mab) mab = d4.z; }
        if (b3 == p) { if (d4.w <= vbits) cle++; else if (d4.w < mab) mab = d4.w; }
      }
    }
    *cntLE = blockSum(cle);
    scan[tid] = mab;
    __syncthreads();
    for (int off = kBT >> 1; off > 0; off >>= 1) {
      if (tid < off) { uint32_t o = scan[tid + off]; if (o < scan[tid]) scan[tid] = o; }
      __syncthreads();
    }
    *minAbove = scan[0];
    __syncthreads();
  };

  // Interpolated quantiles -> IQR bounds (uniform control flow across the block).
  for (int p = 1; p <= kP; ++p) {
    uint32_t n = nv[p];
    float q1, q3;
    if (n == 0) {
      q1 = q3 = 1e9f;   // matches the BIG sentinel path of the reference
    } else {
      float nm1  = (float)(n - 1);
      float pos1 = 0.25f * nm1, pos3 = 0.75f * nm1;
      uint32_t lo1 = (uint32_t)floorf(pos1), hi1 = (uint32_t)ceilf(pos1);
      uint32_t lo3 = (uint32_t)floorf(pos3), hi3 = (uint32_t)ceilf(pos3);
      float f1 = pos1 - floorf(pos1), f3 = pos3 - floorf(pos3);
      uint32_t vlo1 = selectK((uint32_t)p, lo1);
      uint32_t vhi1 = vlo1;
      if (hi1 != lo1) {
        uint32_t cle, mab;
        rankProbe((uint32_t)p, vlo1, &cle, &mab);
        vhi1 = (cle >= hi1 + 1) ? vlo1 : mab;   // handles duplicates exactly
      }
      uint32_t vlo3 = selectK((uint32_t)p, lo3);
      uint32_t vhi3 = vlo3;
      if (hi3 != lo3) {
        uint32_t cle, mab;
        rankProbe((uint32_t)p, vlo3, &cle, &mab);
        vhi3 = (cle >= hi3 + 1) ? vlo3 : mab;
      }
      float a1 = __uint_as_float(vlo1), b1f = __uint_as_float(vhi1);
      float a3 = __uint_as_float(vlo3), b3f = __uint_as_float(vhi3);
      q1 = a1 + (b1f - a1) * f1;
      q3 = a3 + (b3f - a3) * f3;
    }
    if (tid == 0) {
      float iqr = q3 - q1;
      lbA[p] = q1 - 1.5f * iqr;
      ubA[p] = q3 + 1.5f * iqr;
    }
  }
  __syncthreads();

  // ---- Phase 3: raster-order compaction, two persons per traversal -----------
  const float fxv = (float)((double)kW / (2.0 * tan(81.0 * 3.14159265358979323846 / 360.0)));
  const float fyv = (float)((double)kH / (2.0 * tan(59.0 * 3.14159265358979323846 / 360.0)));
  const int i0 = tid * kChunk;
  const int i1 = (i0 + kChunk < kN) ? i0 + kChunk : kN;

  auto compactPair = [&](uint32_t pa, uint32_t pb) {
    const float lba = lbA[pa], uba = ubA[pa];
    const float lbb = (pb <= (uint32_t)kP) ? lbA[pb] : 1.0f;
    const float ubb = (pb <= (uint32_t)kP) ? ubA[pb] : 0.0f;
    uint32_t cA = 0, cB = 0;
    for (int i = i0; i < i1; ++i) {
      uint32_t t = pid8[i];
      if (t == pa) {
        float d = __uint_as_float(dbits[i]);
        cA += (d >= lba && d <= uba);
      } else if (t == pb) {
        float d = __uint_as_float(dbits[i]);
        cB += (d >= lbb && d <= ubb);
      }
    }
    // packed 16-bit-per-person inclusive Hillis-Steele scan (totals <= 30000, no carry)
    scan[tid] = cA | (cB << 16);
    __syncthreads();
    for (int off = 1; off < kBT; off <<= 1) {
      uint32_t v = scan[tid];
      uint32_t a = (tid >= off) ? scan[tid - off] : 0u;
      __syncthreads();
      scan[tid] = v + a;
      __syncthreads();
    }
    uint32_t incl = scan[tid];
    uint32_t last = scan[kBT - 1];
    uint32_t rA = (incl & 0xFFFFu) - cA;       // exclusive prefix = raster rank
    uint32_t rB = (incl >> 16) - cB;
    uint32_t totA = last & 0xFFFFu, totB = last >> 16;
    const int baseA = ((int)pa - 1) * (kK + 1);
    const int baseB = (pb <= (uint32_t)kP) ? ((int)pb - 1) * (kK + 1) : 0;
    for (int i = i0; i < i1; ++i) {
      uint32_t t = pid8[i];
      bool isA = (t == pa), isB = (t == pb);
      if (!(isA | isB)) continue;
      float d  = __uint_as_float(dbits[i]);
      float lb = isA ? lba : lbb;
      float ub = isA ? uba : ubb;
      if (d >= lb && d <= ub) {
        uint32_t r = isA ? rA : rB;
        if (r < (uint32_t)kK) {
          int row = i / kW, col = i - row * kW;
          float xc = ((float)col - (float)kW * 0.5f) / fxv;
          float yc = ((float)row - (float)kH * 0.5f) / fyv;
          int o = (isA ? baseA : baseB) + (int)r;
          ob[o]               = xc * d;
          ob[kStride + o]     = yc * d;
          ob[2 * kStride + o] = d;
        }
        if (isA) ++rA; else ++rB;
      }
    }
    if (tid == 0) {
      ob[baseA + kK] = (totA > 0) ? 1.0f : 0.0f;
      if (pb <= (uint32_t)kP) ob[baseB + kK] = (totB > 0) ? 1.0f : 0.0f;
    }
    __syncthreads();   // protect scan[] reuse
  };
  compactPair(1u, 2u);
  compactPair(3u, 4u);
  compactPair(5u, 0xFFu);
}

extern "C" void kernel_launch(void* const* d_in, const int* in_sizes, int n_in,
                              void* d_out, int out_size, void* d_ws, size_t ws_size,
                              hipStream_t stream) {
  (void)n_in; (void)out_size; (void)d_ws; (void)ws_size;
  const float* in = (const float*)d_in[0];
  float* out = (float*)d_out;
  int B = in_sizes[0] / (3 * kN);
  if (B <= 0) return;
  DepthMask2PointCloudUltra_57243324121545_kernel<<<dim3((unsigned)B), dim3(kBT), 0, stream>>>(in, out);
}